// FP_EMAU_54563264528815
// MI455X (gfx1250) — compile-verified
//
#include <hip/hip_runtime.h>
#include <hip/hip_bf16.h>

#define BB   16
#define CC   256
#define KMU  64
#define NTOT 8500
#define STAGES 3

typedef __attribute__((ext_vector_type(2))) float v2f;
typedef __attribute__((ext_vector_type(8))) float v8f;

__device__ __forceinline__ v8f wmma4(v2f a, v2f b, v8f c) {
  // D = A(16x4) * B(4x16) + C(16x16), fp32
  return __builtin_amdgcn_wmma_f32_16x16x4_f32(false, a, false, b, (short)0, c, false, false);
}

__device__ __forceinline__ v8f vzero8() {
  v8f z = {0.f, 0.f, 0.f, 0.f, 0.f, 0.f, 0.f, 0.f};
  return z;
}

// -------------------- small utility kernels --------------------

__global__ void fill0_kernel(float* __restrict__ p, int n) {
  int i = blockIdx.x * blockDim.x + threadIdx.x;
  if (i < n) p[i] = 0.f;
}

__global__ void copyf_kernel(const float* __restrict__ s, float* __restrict__ d, int n) {
  int i = blockIdx.x * blockDim.x + threadIdx.x;
  if (i < n) d[i] = s[i];
}

__global__ void bcast_mu_kernel(const float* __restrict__ s, float* __restrict__ d) {
  int i = blockIdx.x * blockDim.x + threadIdx.x;   // BB*CC*KMU threads
  d[i] = s[i & (CC * KMU - 1)];                    // CC*KMU = 16384 (pow2)
}

// -------------------- conv1: xt[b,c,off+n] = W1 @ x + b1 --------------------
// grid: (ceil(hw/64), CC/32, BB), block: 32 (one wave -> 32M x 64N tile)
__global__ void conv1_kernel(const float* __restrict__ x, const float* __restrict__ w1,
                             const float* __restrict__ b1, float* __restrict__ xt,
                             int hw, int coff) {
  const int lane = threadIdx.x;
  const int fr = lane & 15;
  const int hh = lane >> 4;
  const int kh = hh << 1;
  const int n0 = blockIdx.x * 64;
  const int m0 = blockIdx.y * 32;
  const int b  = blockIdx.z;
  const float* xb = x + (long)b * CC * hw;
  float* xo = xt + (long)b * CC * NTOT + coff;
  const float* wlo = w1 + (long)(m0 + fr) * CC;
  const float* whi = w1 + (long)(m0 + 16 + fr) * CC;

  // loop-invariant clamped column indices (OOB columns load in-bounds garbage,
  // which is never stored)
  int nc[4];
#pragma unroll
  for (int t = 0; t < 4; ++t) {
    int n = n0 + t * 16 + fr;
    nc[t] = n < hw ? n : hw - 1;
  }

  v8f accL[4], accH[4];
#pragma unroll
  for (int t = 0; t < 4; ++t) { accL[t] = vzero8(); accH[t] = vzero8(); }

#pragma unroll 2
  for (int k0 = 0; k0 < CC; k0 += 4) {
    v2f alo; alo.x = wlo[k0 + kh + 0]; alo.y = wlo[k0 + kh + 1];
    v2f ahi; ahi.x = whi[k0 + kh + 0]; ahi.y = whi[k0 + kh + 1];
    const float* xr0 = xb + (long)(k0 + kh + 0) * hw;
    const float* xr1 = xb + (long)(k0 + kh + 1) * hw;
#pragma unroll
    for (int t = 0; t < 4; ++t) {
      v2f bf; bf.x = xr0[nc[t]]; bf.y = xr1[nc[t]];
      accL[t] = wmma4(alo, bf, accL[t]);
      accH[t] = wmma4(ahi, bf, accH[t]);
    }
  }
#pragma unroll
  for (int t = 0; t < 4; ++t) {
    int n = n0 + t * 16 + fr;
    if (n < hw) {
#pragma unroll
      for (int r = 0; r < 8; ++r) {
        int mL = m0 + r + (hh << 3);
        int mH = mL + 16;
        xo[(long)mL * NTOT + n] = accL[t][r] + b1[mL];
        xo[(long)mH * NTOT + n] = accH[t][r] + b1[mH];
      }
    }
  }
}

// -------------------- EM step 1: z = softmax_k(xt^T @ mu), colsum += z ------
// grid: (ceil(NTOT/16), BB), block: 32; wave computes 16n x 64k logits
__global__ void em_z_kernel(const float* __restrict__ xt, const float* __restrict__ mu,
                            float* __restrict__ z, float* __restrict__ colsum) {
  const int lane = threadIdx.x;
  const int fr = lane & 15;
  const int hh = lane >> 4;
  const int kh = hh << 1;
  const int n0 = blockIdx.x * 16;
  const int b  = blockIdx.y;
  const float* xtb = xt + (long)b * CC * NTOT;
  const float* mub = mu + (long)b * CC * KMU;
  const int na = n0 + fr;
  const int nac = na < NTOT ? na : NTOT - 1;   // clamped, unconditional loads

  v8f a0 = vzero8(), a1 = vzero8(), a2 = vzero8(), a3 = vzero8();
#pragma unroll 2
  for (int c0 = 0; c0 < CC; c0 += 4) {
    v2f a;
    a.x = xtb[(long)(c0 + kh + 0) * NTOT + nac];
    a.y = xtb[(long)(c0 + kh + 1) * NTOT + nac];
    const float* m0p = mub + (long)(c0 + kh + 0) * KMU + fr;
    const float* m1p = mub + (long)(c0 + kh + 1) * KMU + fr;
    v2f b0; b0.x = m0p[0];  b0.y = m1p[0];
    v2f b1; b1.x = m0p[16]; b1.y = m1p[16];
    v2f b2; b2.x = m0p[32]; b2.y = m1p[32];
    v2f b3; b3.x = m0p[48]; b3.y = m1p[48];
    a0 = wmma4(a, b0, a0);
    a1 = wmma4(a, b1, a1);
    a2 = wmma4(a, b2, a2);
    a3 = wmma4(a, b3, a3);
  }

  float ps0 = 0.f, ps1 = 0.f, ps2 = 0.f, ps3 = 0.f;
#pragma unroll
  for (int r = 0; r < 8; ++r) {
    float v0 = a0[r], v1 = a1[r], v2 = a2[r], v3 = a3[r];
    float mx = fmaxf(fmaxf(v0, v1), fmaxf(v2, v3));
    mx = fmaxf(mx, __shfl_xor(mx, 1));
    mx = fmaxf(mx, __shfl_xor(mx, 2));
    mx = fmaxf(mx, __shfl_xor(mx, 4));
    mx = fmaxf(mx, __shfl_xor(mx, 8));
    float e0 = __expf(v0 - mx), e1 = __expf(v1 - mx);
    float e2 = __expf(v2 - mx), e3 = __expf(v3 - mx);
    float s = e0 + e1 + e2 + e3;
    s += __shfl_xor(s, 1);
    s += __shfl_xor(s, 2);
    s += __shfl_xor(s, 4);
    s += __shfl_xor(s, 8);
    float inv = 1.0f / s;
    int nr = n0 + r + (hh << 3);
    bool ok = nr < NTOT;
    float z0 = ok ? e0 * inv : 0.f;
    float z1 = ok ? e1 * inv : 0.f;
    float z2 = ok ? e2 * inv : 0.f;
    float z3 = ok ? e3 * inv : 0.f;
    if (ok) {
      float* zp = z + (long)b * NTOT * KMU + (long)nr * KMU + fr;
      zp[0]  = z0;
      zp[16] = z1;
      zp[32] = z2;
      zp[48] = z3;
    }
    ps0 += z0; ps1 += z1; ps2 += z2; ps3 += z3;
  }
  ps0 += __shfl_xor(ps0, 16);
  ps1 += __shfl_xor(ps1, 16);
  ps2 += __shfl_xor(ps2, 16);
  ps3 += __shfl_xor(ps3, 16);
  if (lane < 16) {
    float* cp = colsum + b * KMU + fr;
    atomicAdd(cp + 0,  ps0);
    atomicAdd(cp + 16, ps1);
    atomicAdd(cp + 32, ps2);
    atomicAdd(cp + 48, ps3);
  }
}

// -------------------- EM step 2: G[b,c,k] += xt @ z (partial over n) --------
// grid: (CC/32, 8, BB), block: 32 ; inner N split into 8 chunks of 1064
__global__ void em_g_kernel(const float* __restrict__ xt, const float* __restrict__ z,
                            float* __restrict__ G) {
  const int lane = threadIdx.x;
  const int fr = lane & 15;
  const int hh = lane >> 4;
  const int kh = hh << 1;
  const int m0 = blockIdx.x * 32;
  const int b  = blockIdx.z;
  int nbeg = blockIdx.y * 1064;
  int nend = nbeg + 1064;
  if (nend > NTOT) nend = NTOT;
  const float* aplo = xt + (long)b * CC * NTOT + (long)(m0 + fr) * NTOT;
  const float* aphi = aplo + (long)16 * NTOT;
  const float* zb = z + (long)b * NTOT * KMU;

  v8f aL0 = vzero8(), aL1 = vzero8(), aL2 = vzero8(), aL3 = vzero8();
  v8f aH0 = vzero8(), aH1 = vzero8(), aH2 = vzero8(), aH3 = vzero8();
#pragma unroll 2
  for (int n = nbeg; n < nend; n += 4) {
    v2f alo; alo.x = aplo[n + kh + 0]; alo.y = aplo[n + kh + 1];
    v2f ahi; ahi.x = aphi[n + kh + 0]; ahi.y = aphi[n + kh + 1];
    const float* z0p = zb + (long)(n + kh + 0) * KMU + fr;
    const float* z1p = zb + (long)(n + kh + 1) * KMU + fr;
    v2f b0; b0.x = z0p[0];  b0.y = z1p[0];
    v2f b1; b1.x = z0p[16]; b1.y = z1p[16];
    v2f b2; b2.x = z0p[32]; b2.y = z1p[32];
    v2f b3; b3.x = z0p[48]; b3.y = z1p[48];
    aL0 = wmma4(alo, b0, aL0);
    aH0 = wmma4(ahi, b0, aH0);
    aL1 = wmma4(alo, b1, aL1);
    aH1 = wmma4(ahi, b1, aH1);
    aL2 = wmma4(alo, b2, aL2);
    aH2 = wmma4(ahi, b2, aH2);
    aL3 = wmma4(alo, b3, aL3);
    aH3 = wmma4(ahi, b3, aH3);
  }
  float* Gb = G + (long)b * CC * KMU;
#pragma unroll
  for (int r = 0; r < 8; ++r) {
    int mL = m0 + r + (hh << 3);
    float* gL = Gb + (long)mL * KMU + fr;
    float* gH = gL + 16 * KMU;
    atomicAdd(gL + 0,  aL0[r]);
    atomicAdd(gL + 16, aL1[r]);
    atomicAdd(gL + 32, aL2[r]);
    atomicAdd(gL + 48, aL3[r]);
    atomicAdd(gH + 0,  aH0[r]);
    atomicAdd(gH + 16, aH1[r]);
    atomicAdd(gH + 32, aH2[r]);
    atomicAdd(gH + 48, aH3[r]);
  }
}

// -------------------- EM step 3: mu = l2norm_c(G / (1e-6 + colsum)) --------
// grid: (KMU, BB), block: 256 (one thread per channel c)
__global__ void mu_norm_kernel(const float* __restrict__ G, const float* __restrict__ colsum,
                               float* __restrict__ muo) {
  const int k = blockIdx.x;
  const int b = blockIdx.y;
  const int c = threadIdx.x;
  float cs = colsum[b * KMU + k];
  float v = G[(long)b * CC * KMU + (long)c * KMU + k] / (1e-6f + cs);
  float s = v * v;
  s += __shfl_xor(s, 1);
  s += __shfl_xor(s, 2);
  s += __shfl_xor(s, 4);
  s += __shfl_xor(s, 8);
  s += __shfl_xor(s, 16);
  __shared__ float red[8];
  if ((threadIdx.x & 31) == 0) red[threadIdx.x >> 5] = s;
  __syncthreads();
  float tot = red[0] + red[1] + red[2] + red[3] + red[4] + red[5] + red[6] + red[7];
  muo[(long)b * CC * KMU + (long)c * KMU + k] = v / (1e-6f + sqrtf(tot));
}

// -------------------- x_rec = relu(mu @ z^T), written over xt ---------------
// grid: (ceil(NTOT/64), CC/32, BB), block: 32
__global__ void xrec_kernel(const float* __restrict__ mu, const float* __restrict__ z,
                            float* __restrict__ xr) {
  const int lane = threadIdx.x;
  const int fr = lane & 15;
  const int hh = lane >> 4;
  const int kh = hh << 1;
  const int n0 = blockIdx.x * 64;
  const int m0 = blockIdx.y * 32;
  const int b  = blockIdx.z;
  const float* mulo = mu + (long)b * CC * KMU + (long)(m0 + fr) * KMU;
  const float* muhi = mulo + 16 * KMU;
  const float* zb   = z + (long)b * NTOT * KMU;

  int nc[4];
#pragma unroll
  for (int t = 0; t < 4; ++t) {
    int n = n0 + t * 16 + fr;
    nc[t] = n < NTOT ? n : NTOT - 1;
  }

  v8f accL[4], accH[4];
#pragma unroll
  for (int t = 0; t < 4; ++t) { accL[t] = vzero8(); accH[t] = vzero8(); }

#pragma unroll 2
  for (int k0 = 0; k0 < KMU; k0 += 4) {
    v2f alo; alo.x = mulo[k0 + kh + 0]; alo.y = mulo[k0 + kh + 1];
    v2f ahi; ahi.x = muhi[k0 + kh + 0]; ahi.y = muhi[k0 + kh + 1];
#pragma unroll
    for (int t = 0; t < 4; ++t) {
      const float* zp = zb + (long)nc[t] * KMU + k0 + kh;
      v2f bf; bf.x = zp[0]; bf.y = zp[1];
      accL[t] = wmma4(alo, bf, accL[t]);
      accH[t] = wmma4(ahi, bf, accH[t]);
    }
  }
  float* xo = xr + (long)b * CC * NTOT;
#pragma unroll
  for (int t = 0; t < 4; ++t) {
    int n = n0 + t * 16 + fr;
    if (n < NTOT) {
#pragma unroll
      for (int r = 0; r < 8; ++r) {
        int mL = m0 + r + (hh << 3);
        int mH = mL + 16;
        xo[(long)mL * NTOT + n] = fmaxf(accL[t][r], 0.f);
        xo[(long)mH * NTOT + n] = fmaxf(accH[t][r], 0.f);
      }
    }
  }
}

// -------------------- conv2 + BN(eval) + residual + relu --------------------
// grid: (ceil(hw/64), CC/32, BB), block: 32
__global__ void conv2_kernel(const float* __restrict__ xr, const float* __restrict__ w2,
                             const float* __restrict__ gamma, const float* __restrict__ beta,
                             const float* __restrict__ mean, const float* __restrict__ var,
                             const float* __restrict__ idn, float* __restrict__ out,
                             int hw, int coff) {
  const int lane = threadIdx.x;
  const int fr = lane & 15;
  const int hh = lane >> 4;
  const int kh = hh << 1;
  const int n0 = blockIdx.x * 64;
  const int m0 = blockIdx.y * 32;
  const int b  = blockIdx.z;
  const float* xb = xr + (long)b * CC * NTOT + coff;
  const float* wlo = w2 + (long)(m0 + fr) * CC;
  const float* whi = w2 + (long)(m0 + 16 + fr) * CC;

  int nc[4];
#pragma unroll
  for (int t = 0; t < 4; ++t) {
    int n = n0 + t * 16 + fr;
    nc[t] = n < hw ? n : hw - 1;
  }

  v8f accL[4], accH[4];
#pragma unroll
  for (int t = 0; t < 4; ++t) { accL[t] = vzero8(); accH[t] = vzero8(); }

#pragma unroll 2
  for (int k0 = 0; k0 < CC; k0 += 4) {
    v2f alo; alo.x = wlo[k0 + kh + 0]; alo.y = wlo[k0 + kh + 1];
    v2f ahi; ahi.x = whi[k0 + kh + 0]; ahi.y = whi[k0 + kh + 1];
    const float* xr0 = xb + (long)(k0 + kh + 0) * NTOT;
    const float* xr1 = xb + (long)(k0 + kh + 1) * NTOT;
#pragma unroll
    for (int t = 0; t < 4; ++t) {
      v2f bf; bf.x = xr0[nc[t]]; bf.y = xr1[nc[t]];
      accL[t] = wmma4(alo, bf, accL[t]);
      accH[t] = wmma4(ahi, bf, accH[t]);
    }
  }
  const float* ib = idn + (long)b * CC * hw;
  float* ob = out + (long)b * CC * hw;
#pragma unroll
  for (int t = 0; t < 4; ++t) {
    int n = n0 + t * 16 + fr;
    if (n < hw) {
#pragma unroll
      for (int r = 0; r < 8; ++r) {
        int mL = m0 + r + (hh << 3);
        int mH = mL + 16;
        float ivL = gamma[mL] * rsqrtf(var[mL] + 1e-5f);
        float shL = beta[mL] - mean[mL] * ivL;
        float yL = accL[t][r] * ivL + shL + ib[(long)mL * hw + n];
        ob[(long)mL * hw + n] = fmaxf(yL, 0.f);
        float ivH = gamma[mH] * rsqrtf(var[mH] + 1e-5f);
        float shH = beta[mH] - mean[mH] * ivH;
        float yH = accH[t][r] * ivH + shH + ib[(long)mH * hw + n];
        ob[(long)mH * hw + n] = fmaxf(yH, 0.f);
      }
    }
  }
}

// -------------------- host-side orchestration --------------------

extern "C" void kernel_launch(void* const* d_in, const int* in_sizes, int n_in,
                              void* d_out, int out_size, void* d_ws, size_t ws_size,
                              hipStream_t stream) {
  (void)in_sizes; (void)n_in; (void)out_size; (void)ws_size;
  const float* x[4] = {(const float*)d_in[0], (const float*)d_in[1],
                       (const float*)d_in[2], (const float*)d_in[3]};
  const float* w1    = (const float*)d_in[4];
  const float* b1    = (const float*)d_in[5];
  const float* w2    = (const float*)d_in[6];
  const float* gamma = (const float*)d_in[7];
  const float* beta  = (const float*)d_in[8];
  const float* mean  = (const float*)d_in[9];
  const float* var   = (const float*)d_in[10];
  const float* mu_in = (const float*)d_in[11];
  float* outp = (float*)d_out;

  float* ws  = (float*)d_ws;
  float* xt  = ws;                                  // BB*CC*NTOT  (reused for x_rec)
  float* zz  = xt  + (size_t)BB * CC * NTOT;        // BB*NTOT*KMU
  float* muA = zz  + (size_t)BB * NTOT * KMU;       // BB*CC*KMU
  float* muB = muA + (size_t)BB * CC * KMU;         // BB*CC*KMU
  float* G   = muB + (size_t)BB * CC * KMU;         // BB*CC*KMU
  float* cs  = G   + (size_t)BB * CC * KMU;         // BB*KMU

  const int hw[4]  = {6400, 1600, 400, 100};
  const int off[4] = {0, 6400, 8000, 8400};

  // mu broadcast to all batches
  bcast_mu_kernel<<<dim3((BB * CC * KMU) / 256), 256, 0, stream>>>(mu_in, muA);

  // conv1 for each scale -> xt
  for (int i = 0; i < 4; ++i) {
    conv1_kernel<<<dim3((hw[i] + 63) / 64, CC / 32, BB), 32, 0, stream>>>(
        x[i], w1, b1, xt, hw[i], off[i]);
  }

  float* cur = muA;
  float* nxt = muB;
  for (int s = 0; s < STAGES; ++s) {
    fill0_kernel<<<dim3((BB * KMU + 255) / 256), 256, 0, stream>>>(cs, BB * KMU);
    fill0_kernel<<<dim3((BB * CC * KMU + 255) / 256), 256, 0, stream>>>(G, BB * CC * KMU);
    em_z_kernel<<<dim3((NTOT + 15) / 16, BB), 32, 0, stream>>>(xt, cur, zz, cs);
    em_g_kernel<<<dim3(CC / 32, 8, BB), 32, 0, stream>>>(xt, zz, G);
    mu_norm_kernel<<<dim3(KMU, BB), 256, 0, stream>>>(G, cs, nxt);
    float* t = cur; cur = nxt; nxt = t;
  }

  // x_rec = relu(mu @ z^T) overwrites xt
  xrec_kernel<<<dim3((NTOT + 63) / 64, CC / 32, BB), 32, 0, stream>>>(cur, zz, xt);

  // conv2 + BN + residual + relu per scale
  size_t oo = 0;
  for (int i = 0; i < 4; ++i) {
    conv2_kernel<<<dim3((hw[i] + 63) / 64, CC / 32, BB), 32, 0, stream>>>(
        xt, w2, gamma, beta, mean, var, x[i], outp + oo, hw[i], off[i]);
    oo += (size_t)BB * CC * hw[i];
  }

  // final mu to output tail
  copyf_kernel<<<dim3((BB * CC * KMU + 255) / 256), 256, 0, stream>>>(cur, outp + oo, BB * CC * KMU);
}